// FullTransformer_4715874091909
// MI455X (gfx1250) — compile-verified
//
#include <hip/hip_runtime.h>
#include <math.h>

// ---------------------------------------------------------------------------
// CDNA5 fp32 WMMA helpers: D = A(16x4) * B(4x16) + C(16x16), full fp32.
// A frag: lane -> M=lane%16 ; component v + half hh -> K = 2*hh + v
// B frag: lane -> N=lane%16 ; component v + half hh -> K = 2*hh + v
// C/D   : VGPR r + half hh -> M = r + 8*hh ; lane -> N = lane%16
// ---------------------------------------------------------------------------
typedef __attribute__((ext_vector_type(2))) float v2f;
typedef __attribute__((ext_vector_type(8))) float v8f;

__device__ __forceinline__ v8f wmma4(v2f a, v2f b, v8f c) {
  return __builtin_amdgcn_wmma_f32_16x16x4_f32(
      /*neg_a=*/false, a, /*neg_b=*/false, b,
      /*c_mod=*/(short)0, c, /*reuse_a=*/false, /*reuse_b=*/false);
}

// ---------------------------------------------------------------------------
// Generic GEMM:  C[N x NC] = A[N x K] @ W[K x NC] + bias, optional epilogue.
// One wave per block; each wave computes a 64-row x 64-col tile
// (4 row-tiles x 4 col-tiles) so each B fragment feeds 4 WMMAs:
// per K-step: 4x b64 A-loads + 8x b32 B-loads -> 16 WMMAs.
// EPI (compile-time): 0 = none, 1 = exact GELU, 2 = add residual
// (resid stride == NC). Compile-time EPI keeps the epilogue branch-free so
// residual loads can be claused/overlapped instead of serialized.
// ---------------------------------------------------------------------------
template <int EPI>
__global__ __launch_bounds__(32) void gemm_bias_kernel(
    const float* __restrict__ A, const float* __restrict__ W,
    const float* __restrict__ bias, const float* __restrict__ resid,
    float* __restrict__ C, int K, int NC) {
  const int lane = threadIdx.x;
  const int m = lane & 15;
  const int hh = lane >> 4;
  const int row0 = blockIdx.x * 64;
  const int n0 = blockIdx.y * 64;

  v8f acc[4][4];
#pragma unroll
  for (int rt = 0; rt < 4; ++rt)
#pragma unroll
    for (int t = 0; t < 4; ++t) acc[rt][t] = {};

  for (int kb = 0; kb < K; kb += 4) {
    v2f a[4];
#pragma unroll
    for (int rt = 0; rt < 4; ++rt)
      a[rt] = *(const v2f*)(A + (size_t)(row0 + rt * 16 + m) * K + kb + hh * 2);
    v2f b[4];
#pragma unroll
    for (int t = 0; t < 4; ++t) {
      const int n = n0 + t * 16 + m;
      b[t].x = W[(size_t)(kb + hh * 2) * NC + n];
      b[t].y = W[(size_t)(kb + hh * 2 + 1) * NC + n];
    }
#pragma unroll
    for (int rt = 0; rt < 4; ++rt)
#pragma unroll
      for (int t = 0; t < 4; ++t) acc[rt][t] = wmma4(a[rt], b[t], acc[rt][t]);
  }

  // ---- epilogue (straight-line; EPI is compile-time) ------------------------
  float bn[4];
#pragma unroll
  for (int t = 0; t < 4; ++t) bn[t] = bias[n0 + t * 16 + m];

  float res[4][4][8];
  if (EPI == 2) {
#pragma unroll
    for (int rt = 0; rt < 4; ++rt)
#pragma unroll
      for (int t = 0; t < 4; ++t)
#pragma unroll
        for (int r = 0; r < 8; ++r)
          res[rt][t][r] = resid[(size_t)(row0 + rt * 16 + r + 8 * hh) * NC +
                                n0 + t * 16 + m];
  }

#pragma unroll
  for (int rt = 0; rt < 4; ++rt)
#pragma unroll
    for (int t = 0; t < 4; ++t)
#pragma unroll
      for (int r = 0; r < 8; ++r) {
        const int row = row0 + rt * 16 + r + 8 * hh;
        const int col = n0 + t * 16 + m;
        float v = acc[rt][t][r] + bn[t];
        if (EPI == 1) v = 0.5f * v * (1.0f + erff(v * 0.70710678118654752f));
        if (EPI == 2) v += res[rt][t][r];
        C[(size_t)row * NC + col] = v;
      }
}

// ---------------------------------------------------------------------------
// Optional residual-add + LayerNorm over hid=256, one wave per token.
// If ADD: h[tok] += addend[tok] is written back in place.
// ln_out may alias addend (each thread reads its indices before writing them).
// ---------------------------------------------------------------------------
template <bool ADD>
__global__ __launch_bounds__(32) void add_ln_kernel(
    float* __restrict__ h, const float* __restrict__ addend,
    const float* __restrict__ g, const float* __restrict__ beta,
    float* __restrict__ ln_out) {
  const int lane = threadIdx.x;
  const size_t base = (size_t)blockIdx.x * 256;

  float v[8];
  float s = 0.f, ss = 0.f;
#pragma unroll
  for (int i = 0; i < 8; ++i) {
    const int c = i * 32 + lane;
    float x = h[base + c];
    if (ADD) x += addend[base + c];
    v[i] = x;
    s += x;
    ss += x * x;
  }
  if (ADD) {
#pragma unroll
    for (int i = 0; i < 8; ++i) h[base + i * 32 + lane] = v[i];
  }
  for (int off = 16; off; off >>= 1) {
    s += __shfl_xor(s, off);
    ss += __shfl_xor(ss, off);
  }
  const float mu = s * (1.0f / 256.0f);
  const float var = ss * (1.0f / 256.0f) - mu * mu;
  const float rs = rsqrtf(var + 1e-5f);
#pragma unroll
  for (int i = 0; i < 8; ++i) {
    const int c = i * 32 + lane;
    ln_out[base + c] = (v[i] - mu) * rs * g[c] + beta[c];
  }
}

// ---------------------------------------------------------------------------
// Windowed causal attention. qkv is [N x 768] (q|k|v, 256 each), one wave per
// 16-query block. Window: 0 <= q-k <= 128 within each batch of T=4096 tokens,
// so a query block sees at most 144 keys = 9 key-tiles of 16.
// Scores and P@V both via fp32 WMMA; P goes through LDS to switch from the
// C register layout to the A register layout.
// ---------------------------------------------------------------------------
#define ATT_T 4096
#define ATT_QS 768
#define ATT_WIN 128
#define ATT_PST 148  // padded LDS row stride (floats)

__global__ __launch_bounds__(32) void attn_kernel(const float* __restrict__ qkv,
                                                  float* __restrict__ O) {
  __shared__ float P[16 * ATT_PST];
  const int lane = threadIdx.x;
  const int m = lane & 15;
  const int hh = lane >> 4;
  const int q0 = blockIdx.x * 16;
  const int bb = q0 & ~(ATT_T - 1);  // batch start token

  const float* Q = qkv;
  const float* Km = qkv + 256;
  const float* V = qkv + 512;

  // ---- scores: S[16 x 144] = Q_tile @ K_tile^T ------------------------------
  v8f s[9];
#pragma unroll
  for (int t = 0; t < 9; ++t) s[t] = {};

  for (int d = 0; d < 256; d += 4) {
    const v2f a = *(const v2f*)(Q + (size_t)(q0 + m) * ATT_QS + d + hh * 2);
#pragma unroll
    for (int t = 0; t < 9; ++t) {
      const int key = q0 - ATT_WIN + t * 16 + m;
      const int kc = key < bb ? bb : key;  // clamp; masked to -inf below
      const v2f b = *(const v2f*)(Km + (size_t)kc * ATT_QS + d + hh * 2);
      s[t] = wmma4(a, b, s[t]);
    }
  }

  // ---- mask + softmax (per-row reductions across 16-lane halves) ------------
  const float NEG = -3.402823466e38f;
  float rmax[8], rsum[8];
#pragma unroll
  for (int r = 0; r < 8; ++r) rmax[r] = NEG;
#pragma unroll
  for (int t = 0; t < 9; ++t) {
    const int key = q0 - ATT_WIN + t * 16 + m;
#pragma unroll
    for (int r = 0; r < 8; ++r) {
      const int q = q0 + r + 8 * hh;
      const int d = q - key;
      const bool ok = (key >= bb) && (d >= 0) && (d <= ATT_WIN);
      const float val = ok ? s[t][r] * 0.0625f : NEG;  // scale = 1/sqrt(256)
      s[t][r] = val;
      rmax[r] = fmaxf(rmax[r], val);
    }
  }
#pragma unroll
  for (int r = 0; r < 8; ++r)
    for (int off = 1; off < 16; off <<= 1)
      rmax[r] = fmaxf(rmax[r], __shfl_xor(rmax[r], off));
#pragma unroll
  for (int r = 0; r < 8; ++r) rsum[r] = 0.f;
#pragma unroll
  for (int t = 0; t < 9; ++t)
#pragma unroll
    for (int r = 0; r < 8; ++r) {
      const float e = expf(s[t][r] - rmax[r]);
      s[t][r] = e;
      rsum[r] += e;
    }
#pragma unroll
  for (int r = 0; r < 8; ++r)
    for (int off = 1; off < 16; off <<= 1) rsum[r] += __shfl_xor(rsum[r], off);

#pragma unroll
  for (int t = 0; t < 9; ++t)
#pragma unroll
    for (int r = 0; r < 8; ++r)
      P[(r + 8 * hh) * ATT_PST + t * 16 + m] = s[t][r] / rsum[r];

  __syncthreads();  // single-wave WG: DS ordering is in-order anyway

  // ---- out = P[16 x 144] @ V[144 x 256], two 128-col halves -----------------
#pragma unroll
  for (int ch = 0; ch < 2; ++ch) {
    const int n0 = ch * 128;
    v8f o[8];
#pragma unroll
    for (int tc = 0; tc < 8; ++tc) o[tc] = {};
    for (int t = 0; t < 9; ++t) {
      const int kt0 = q0 - ATT_WIN + t * 16;
#pragma unroll
      for (int kk = 0; kk < 16; kk += 4) {
        // A frag from LDS: row=m, K = kk + 2*hh + v  (contiguous b64 read)
        const v2f a = *(const v2f*)&P[m * ATT_PST + t * 16 + kk + hh * 2];
        const int k0 = kt0 + kk + hh * 2;
        const int kcx = k0 < bb ? bb : k0;  // clamped rows have P==0
        const int kcy = (k0 + 1) < bb ? bb : (k0 + 1);
#pragma unroll
        for (int tc = 0; tc < 8; ++tc) {
          const int col = n0 + tc * 16 + m;
          v2f b;
          b.x = V[(size_t)kcx * ATT_QS + col];
          b.y = V[(size_t)kcy * ATT_QS + col];
          o[tc] = wmma4(a, b, o[tc]);
        }
      }
    }
#pragma unroll
    for (int tc = 0; tc < 8; ++tc)
#pragma unroll
      for (int r = 0; r < 8; ++r)
        O[(size_t)(q0 + r + 8 * hh) * 256 + n0 + tc * 16 + m] = o[tc][r];
  }
}

// ---------------------------------------------------------------------------
// Host-side pipeline. Workspace layout (floats), N = 16384 tokens:
//   [0, 256N)        h        (residual stream, updated in place)
//   [256N, 512N)     ln / attn_out  (time-multiplexed; safe per-element)
//   [512N, 1280N)    qkv      (later reused as m1)
// Total 1280*N*4 = 80 MB.
// ---------------------------------------------------------------------------
extern "C" void kernel_launch(void* const* d_in, const int* in_sizes, int n_in,
                              void* d_out, int out_size, void* d_ws,
                              size_t ws_size, hipStream_t stream) {
  const float* x = (const float*)d_in[0];
  const float* W_in = (const float*)d_in[1];
  const float* b_in = (const float*)d_in[2];
  const float* W_qkv = (const float*)d_in[3];
  const float* b_qkv = (const float*)d_in[4];
  const float* g_attn = (const float*)d_in[5];
  const float* be_attn = (const float*)d_in[6];
  const float* g_mlp = (const float*)d_in[7];
  const float* be_mlp = (const float*)d_in[8];
  const float* W1 = (const float*)d_in[9];
  const float* b1 = (const float*)d_in[10];
  const float* W2 = (const float*)d_in[11];
  const float* b2 = (const float*)d_in[12];
  const float* W_out = (const float*)d_in[13];
  const float* b_out = (const float*)d_in[14];
  float* out = (float*)d_out;

  const int N = in_sizes[0] / 64;  // tokens = B*T = 16384

  float* ws = (float*)d_ws;
  float* h = ws;
  float* ln = ws + (size_t)N * 256;
  float* attn = ln;  // reuse: ln1 dead once qkv exists; rewritten as ln2 later
  float* qkv = ws + (size_t)N * 512;
  float* m1 = qkv;  // reuse: qkv dead after attention

  const dim3 blk(32);
  const int RB64 = N / 64;  // 64-row GEMM blocks
  const int QB = N / 16;    // 16-query attention blocks

  // 1) h = x @ W_in + b_in                  (K=64, NC=256)
  gemm_bias_kernel<0><<<dim3(RB64, 4), blk, 0, stream>>>(x, W_in, b_in,
                                                         nullptr, h, 64, 256);
  // 2) ln1 = LN(h)
  add_ln_kernel<false>
      <<<dim3(N), blk, 0, stream>>>(h, nullptr, g_attn, be_attn, ln);
  // 3) qkv = ln1 @ W_qkv + b_qkv            (K=256, NC=768)
  gemm_bias_kernel<0><<<dim3(RB64, 12), blk, 0, stream>>>(
      ln, W_qkv, b_qkv, nullptr, qkv, 256, 768);
  // 4) attn_out = softmax(windowed QK^T) @ V
  attn_kernel<<<dim3(QB), blk, 0, stream>>>(qkv, attn);
  // 5) h += attn_out ; ln2 = LN(h)
  add_ln_kernel<true>
      <<<dim3(N), blk, 0, stream>>>(h, attn, g_mlp, be_mlp, ln);
  // 6) m1 = gelu(ln2 @ W1 + b1)
  gemm_bias_kernel<1><<<dim3(RB64, 4), blk, 0, stream>>>(ln, W1, b1, nullptr,
                                                         m1, 256, 256);
  // 7) h = h + m1 @ W2 + b2
  gemm_bias_kernel<2><<<dim3(RB64, 4), blk, 0, stream>>>(m1, W2, b2, h, h, 256,
                                                         256);
  // 8) out = h @ W_out + b_out              (K=256, NC=64)
  gemm_bias_kernel<0><<<dim3(RB64, 1), blk, 0, stream>>>(h, W_out, b_out,
                                                         nullptr, out, 256, 64);
}